// GroupedESN_72782515799007
// MI455X (gfx1250) — compile-verified
//
#include <hip/hip_runtime.h>
#include <math.h>

// ---- problem constants (match reference) ----
constexpr int kE = 4;      // num ESNs
constexpr int kD = 64;     // input dim
constexpr int kR = 1024;   // reservoir dim
constexpr int kB = 16;     // batch
constexpr int kT = 1024;   // timesteps
constexpr float kLeak = 0.3f;

constexpr int kSlices   = 16;              // workgroups per ESN
constexpr int kRowsWG   = kR / kSlices;    // 64 reservoir rows per WG
constexpr int kThreads  = 128;             // 4 waves (wave32)
constexpr int kBlocks   = kE * kSlices;    // 64 persistent WGs
constexpr int kKtot     = kR + kD;         // 1088 (W ++ W_in columns)
constexpr int kLdsFloats = (kKtot / 2) * (2 * kRowsWG); // 544*128 = 69632 (278.5 KB)
constexpr int kChunk    = 64;              // K values staged per burst (16 WMMAs)

typedef __attribute__((ext_vector_type(2))) float v2f;
typedef __attribute__((ext_vector_type(8))) float v8f;

__global__ void esn_bar_init(unsigned* bar) {
    if (threadIdx.x < 2) bar[threadIdx.x] = 0u;
}

__device__ __forceinline__ void grid_barrier(unsigned* cnt, unsigned* phase, unsigned target) {
    __threadfence();          // release my h stores to device scope
    __syncthreads();
    if (threadIdx.x == 0) {
        unsigned prev = __hip_atomic_fetch_add(cnt, 1u, __ATOMIC_ACQ_REL,
                                               __HIP_MEMORY_SCOPE_AGENT);
        if (prev == (unsigned)(kBlocks - 1)) {
            __hip_atomic_store(cnt, 0u, __ATOMIC_RELAXED, __HIP_MEMORY_SCOPE_AGENT);
            __hip_atomic_store(phase, target, __ATOMIC_RELEASE, __HIP_MEMORY_SCOPE_AGENT);
        } else {
            while (__hip_atomic_load(phase, __ATOMIC_ACQUIRE,
                                     __HIP_MEMORY_SCOPE_AGENT) < target) {
                __builtin_amdgcn_s_sleep(2);
            }
        }
    }
    __syncthreads();
    __threadfence();          // acquire: invalidate before reading peers' h
}

__global__ __launch_bounds__(kThreads, 1)
void esn_persistent(const float* __restrict__ x,      // [B][T][D]
                    const float* __restrict__ Win,    // [E][R][D]
                    const float* __restrict__ W,      // [E][R][R]
                    float* __restrict__ out,          // [B][E*R]
                    float* __restrict__ hbuf,         // 2 * [E][B][R] ping-pong
                    unsigned* __restrict__ bar) {
    extern __shared__ float lds[];  // pair-interleaved W^T slice:
                                    // lds[(j>>1)*128 + i*2 + (j&1)] = Wrow[i][j]
    const int blk   = blockIdx.x;
    const int e     = blk >> 4;          // / kSlices
    const int slice = blk & 15;
    const int nbase = slice * kRowsWG;
    const int tid   = threadIdx.x;
    const int wave  = tid >> 5;
    const int lane  = tid & 31;
    const int kh    = lane >> 4;         // K-half (0: K,K+1 ; 1: K+2,K+3)
    const int lm    = lane & 15;         // M row (A) / N col (B,C,D)

    // ---- one-time LDS preload: W slice (64x1024) + Win slice (64x64) ----
    const float* We  = W   + (size_t)e * kR * kR + (size_t)nbase * kR;
    const float* Wie = Win + (size_t)e * kR * kD + (size_t)nbase * kD;
    for (int m = tid; m < kRowsWG * kR; m += kThreads) {
        int i = m >> 10, j = m & (kR - 1);                 // coalesced over j
        lds[(j >> 1) * (2 * kRowsWG) + (i << 1) + (j & 1)] = We[(size_t)i * kR + j];
    }
    for (int m = tid; m < kRowsWG * kD; m += kThreads) {
        int i = m >> 6, d = m & (kD - 1);
        int j = kR + d;
        lds[(j >> 1) * (2 * kRowsWG) + (i << 1) + (j & 1)] = Wie[i * kD + d];
    }

    // ---- zero my slice of h ping buffer ----
    float* h0 = hbuf;
    float* h1 = hbuf + (size_t)kE * kB * kR;
    {
        float* hE0 = h0 + (size_t)e * kB * kR;
        for (int m = tid; m < kB * kRowsWG; m += kThreads) {
            int b = m >> 6, i = m & (kRowsWG - 1);
            hE0[(size_t)b * kR + nbase + i] = 0.0f;
        }
    }
    __syncthreads();
    unsigned epoch = 1;
    grid_barrier(&bar[0], &bar[1], epoch++);   // h0 fully zeroed everywhere

    // per-wave constants
    const int n_lds = wave * 16 + lm;              // B/N column within slice
    const int nglob = nbase + n_lds;               // global reservoir row index
    const float* ldsB0 = &lds[(size_t)kh * (2 * kRowsWG) + n_lds * 2];
    const float* ldsBx = &lds[((size_t)(kR / 2) + kh) * (2 * kRowsWG) + n_lds * 2];

    for (int t = 0; t < kT; ++t) {
        const float* hold = ((t & 1) ? h1 : h0) + (size_t)e * kB * kR;
        float*       hnew = ((t & 1) ? h0 : h1) + (size_t)e * kB * kR;

        v8f acc0 = {}; v8f acc1 = {};

        // A operand base: h[b=lm][k + 2*kh] (64-bit loads)
        const float* aRow = hold + (size_t)lm * kR + 2 * kh;
        const float* xRow = x + ((size_t)lm * kT + t) * kD + 2 * kh;
        if (t + 1 < kT)
            __builtin_prefetch(xRow + kD, 0, 1);   // next timestep's x -> global_prefetch

        // K over reservoir state: stage a 64-wide K chunk (16 A/B pairs) into
        // registers, then fire 16 back-to-back WMMAs; unroll 2 so the next
        // chunk's load clause overlaps the current WMMA burst.
        #pragma unroll 2
        for (int kc = 0; kc < kR; kc += kChunk) {
            v2f a[16], b[16];
            #pragma unroll
            for (int u = 0; u < 16; ++u) {
                a[u] = *(const v2f*)(aRow + kc + 4 * u);
                b[u] = *(const v2f*)(ldsB0 + (size_t)((kc >> 1) + 2 * u) * (2 * kRowsWG));
            }
            #pragma unroll
            for (int u = 0; u < 16; ++u) {
                if (u & 1)
                    acc1 = __builtin_amdgcn_wmma_f32_16x16x4_f32(
                               false, a[u], false, b[u], (short)0, acc1, false, false);
                else
                    acc0 = __builtin_amdgcn_wmma_f32_16x16x4_f32(
                               false, a[u], false, b[u], (short)0, acc0, false, false);
            }
        }
        // K over input drive (64 cols = fused W_in * x_t), same staged pattern
        {
            v2f a[16], b[16];
            #pragma unroll
            for (int u = 0; u < 16; ++u) {
                a[u] = *(const v2f*)(xRow + 4 * u);
                b[u] = *(const v2f*)(ldsBx + (size_t)(2 * u) * (2 * kRowsWG));
            }
            #pragma unroll
            for (int u = 0; u < 16; ++u) {
                if (u & 1)
                    acc1 = __builtin_amdgcn_wmma_f32_16x16x4_f32(
                               false, a[u], false, b[u], (short)0, acc1, false, false);
                else
                    acc0 = __builtin_amdgcn_wmma_f32_16x16x4_f32(
                               false, a[u], false, b[u], (short)0, acc0, false, false);
            }
        }

        // epilogue: h' = (1-a)*h + a*tanh(pre); C/D layout: b = v + 8*kh, n = lm
        const int bbase = kh * 8;
        #pragma unroll
        for (int v = 0; v < 8; ++v) {
            float pre = acc0[v] + acc1[v];
            int   b   = bbase + v;
            float ho  = hold[(size_t)b * kR + nglob];
            float hn  = (1.0f - kLeak) * ho + kLeak * tanhf(pre);
            if (t == kT - 1)
                out[(size_t)b * (kE * kR) + (size_t)e * kR + nglob] = hn;
            else
                hnew[(size_t)b * kR + nglob] = hn;
        }

        if (t != kT - 1)
            grid_barrier(&bar[0], &bar[1], epoch++);
    }
}

extern "C" void kernel_launch(void* const* d_in, const int* in_sizes, int n_in,
                              void* d_out, int out_size, void* d_ws, size_t ws_size,
                              hipStream_t stream) {
    const float* x   = (const float*)d_in[0];   // [B,T,D]
    const float* Win = (const float*)d_in[1];   // [E,R,D]
    const float* W   = (const float*)d_in[2];   // [E,R,R]
    float*       out = (float*)d_out;           // [B, E*R]

    float*    hbuf = (float*)d_ws;                                   // 512 KB ping-pong
    unsigned* bar  = (unsigned*)((char*)d_ws +
                       (size_t)2 * kE * kB * kR * sizeof(float));    // 2 words

    esn_bar_init<<<1, 64, 0, stream>>>(bar);
    const size_t ldsBytes = (size_t)kLdsFloats * sizeof(float);      // 278528 B
    esn_persistent<<<kBlocks, kThreads, ldsBytes, stream>>>(x, Win, W, out, hbuf, bar);
}